// GLIF3Triton_27925877359138
// MI455X (gfx1250) — compile-verified
//
#include <hip/hip_runtime.h>

// GLIF3 neuron update: purely elementwise over B=8388608 neurons, M=2.
// Memory-bound (72 B/neuron, ~0.28 FLOP/B) -> optimize for 23.3 TB/s HBM:
// 128-bit coalesced non-temporal loads/stores, 4 neurons/thread, wave32.

typedef float f4 __attribute__((ext_vector_type(4)));

#define GLIF_DT 1.0f   // reference DT

struct GlifOut {
    float vpost, Ip0, Ip1, s;
};

__device__ __forceinline__ GlifOut glif3_one(
    float v, float x, float vth, float vres, float vrest, float cm, float tau,
    float I0, float I1, float k0, float k1, float aa0, float aa1, float nr)
{
    GlifOut o;
    // a = exp(-DT/tau); membrane decay
    float a      = __expf(-GLIF_DT / tau);
    float Isum   = I0 + I1;
    float vinf   = vrest + tau * (x + Isum) / cm;
    float vprime = vinf + (v - vinf) * a;
    // spike = (v' - v_th > 0) * not_refrac   (not_refrac is already 0/1)
    o.s = ((vprime - vth) > 0.0f) ? nr : 0.0f;
    // soft reset (HARD_RESET == False)
    o.vpost = vprime - (vth - vres) * o.s;
    // ascending currents: Iasc * exp(-k*DT) + asc_amps * s
    o.Ip0 = I0 * __expf(-k0 * GLIF_DT) + aa0 * o.s;
    o.Ip1 = I1 * __expf(-k1 * GLIF_DT) + aa1 * o.s;
    return o;
}

__global__ __launch_bounds__(256) void glif3_kernel(
    const f4* __restrict__ v,          // B/4
    const f4* __restrict__ Iasc,       // 2*B/4  (interleaved pairs, row-major (B,2))
    const f4* __restrict__ x,
    const f4* __restrict__ v_th,
    const f4* __restrict__ v_reset,
    const f4* __restrict__ v_rest,
    const f4* __restrict__ c_m,
    const f4* __restrict__ tau,
    const f4* __restrict__ k,          // 2*B/4
    const f4* __restrict__ asc_amps,   // 2*B/4
    const f4* __restrict__ not_refrac,
    f4* __restrict__ v_post,           // B/4
    f4* __restrict__ I_post,           // 2*B/4
    f4* __restrict__ s_out,            // B/4
    int n4)                            // B/4
{
    int i = blockIdx.x * blockDim.x + threadIdx.x;
    if (i >= n4) return;

    // 128-bit non-temporal streaming loads (dataset >> 192MB L2; no reuse)
    f4 vv   = __builtin_nontemporal_load(&v[i]);
    f4 xx   = __builtin_nontemporal_load(&x[i]);
    f4 vt   = __builtin_nontemporal_load(&v_th[i]);
    f4 vr   = __builtin_nontemporal_load(&v_reset[i]);
    f4 vrs  = __builtin_nontemporal_load(&v_rest[i]);
    f4 cm   = __builtin_nontemporal_load(&c_m[i]);
    f4 tu   = __builtin_nontemporal_load(&tau[i]);
    f4 nr   = __builtin_nontemporal_load(&not_refrac[i]);
    // (B,2) arrays: 4 neurons -> two contiguous float4s
    f4 Ia0  = __builtin_nontemporal_load(&Iasc[2 * i + 0]);     // n0:{I0,I1} n1:{I0,I1}
    f4 Ia1  = __builtin_nontemporal_load(&Iasc[2 * i + 1]);     // n2:{I0,I1} n3:{I0,I1}
    f4 kk0  = __builtin_nontemporal_load(&k[2 * i + 0]);
    f4 kk1  = __builtin_nontemporal_load(&k[2 * i + 1]);
    f4 am0  = __builtin_nontemporal_load(&asc_amps[2 * i + 0]);
    f4 am1  = __builtin_nontemporal_load(&asc_amps[2 * i + 1]);

    f4 vp, sp, Ip0, Ip1;

    // neuron 0  (Iasc pair in Ia0[0..1])
    {
        GlifOut o = glif3_one(vv[0], xx[0], vt[0], vr[0], vrs[0], cm[0], tu[0],
                              Ia0[0], Ia0[1], kk0[0], kk0[1], am0[0], am0[1], nr[0]);
        vp[0] = o.vpost; sp[0] = o.s; Ip0[0] = o.Ip0; Ip0[1] = o.Ip1;
    }
    // neuron 1  (Iasc pair in Ia0[2..3])
    {
        GlifOut o = glif3_one(vv[1], xx[1], vt[1], vr[1], vrs[1], cm[1], tu[1],
                              Ia0[2], Ia0[3], kk0[2], kk0[3], am0[2], am0[3], nr[1]);
        vp[1] = o.vpost; sp[1] = o.s; Ip0[2] = o.Ip0; Ip0[3] = o.Ip1;
    }
    // neuron 2  (Iasc pair in Ia1[0..1])
    {
        GlifOut o = glif3_one(vv[2], xx[2], vt[2], vr[2], vrs[2], cm[2], tu[2],
                              Ia1[0], Ia1[1], kk1[0], kk1[1], am1[0], am1[1], nr[2]);
        vp[2] = o.vpost; sp[2] = o.s; Ip1[0] = o.Ip0; Ip1[1] = o.Ip1;
    }
    // neuron 3  (Iasc pair in Ia1[2..3])
    {
        GlifOut o = glif3_one(vv[3], xx[3], vt[3], vr[3], vrs[3], cm[3], tu[3],
                              Ia1[2], Ia1[3], kk1[2], kk1[3], am1[2], am1[3], nr[3]);
        vp[3] = o.vpost; sp[3] = o.s; Ip1[2] = o.Ip0; Ip1[3] = o.Ip1;
    }

    __builtin_nontemporal_store(vp,  &v_post[i]);
    __builtin_nontemporal_store(Ip0, &I_post[2 * i + 0]);
    __builtin_nontemporal_store(Ip1, &I_post[2 * i + 1]);
    __builtin_nontemporal_store(sp,  &s_out[i]);
}

extern "C" void kernel_launch(void* const* d_in, const int* in_sizes, int n_in,
                              void* d_out, int out_size, void* d_ws, size_t ws_size,
                              hipStream_t stream) {
    (void)n_in; (void)out_size; (void)d_ws; (void)ws_size;

    // setup_inputs() dict order
    const f4* v          = (const f4*)d_in[0];
    const f4* Iasc       = (const f4*)d_in[1];
    const f4* x          = (const f4*)d_in[2];
    const f4* v_th       = (const f4*)d_in[3];
    const f4* v_reset    = (const f4*)d_in[4];
    const f4* v_rest     = (const f4*)d_in[5];
    const f4* c_m        = (const f4*)d_in[6];
    const f4* tau        = (const f4*)d_in[7];
    const f4* k          = (const f4*)d_in[8];
    const f4* asc_amps   = (const f4*)d_in[9];
    const f4* not_refrac = (const f4*)d_in[10];

    const int B  = in_sizes[0];      // 8388608, divisible by 4
    const int n4 = B / 4;

    // d_out = [v_post (B) | I_post (2B) | s (B)] as float32, flat in return order
    float* out = (float*)d_out;
    f4* v_post = (f4*)(out);
    f4* I_post = (f4*)(out + (size_t)B);
    f4* s_out  = (f4*)(out + (size_t)3 * B);

    const int block = 256;                      // 8 wave32 waves per block
    const int grid  = (n4 + block - 1) / block; // 8192 blocks

    glif3_kernel<<<grid, block, 0, stream>>>(
        v, Iasc, x, v_th, v_reset, v_rest, c_m, tau, k, asc_amps, not_refrac,
        v_post, I_post, s_out, n4);
}